// GCNConv_90924457656718
// MI455X (gfx1250) — compile-verified
//
#include <hip/hip_runtime.h>

typedef __attribute__((ext_vector_type(2))) float v2f;
typedef __attribute__((ext_vector_type(8))) float v8f;

#define DIM 64
#define WAVES_PER_BLOCK 8

// ---------------------------------------------------------------------------
// Kernel 1: zero the output (atomics accumulate, and graph replays re-run us)
// ---------------------------------------------------------------------------
__global__ void zero_f4(float4* __restrict__ p, int n4) {
    int i = blockIdx.x * blockDim.x + threadIdx.x;
    if (i < n4) p[i] = make_float4(0.f, 0.f, 0.f, 0.f);
}

// ---------------------------------------------------------------------------
// Kernel 2: H = X @ W  via V_WMMA_F32_16X16X4_F32 (full f32 precision)
// One wave -> 16 rows x 64 cols of H. K loop: 16 steps of 4.
//
// A-frag (16x4, MxK): lane L: M=L&15; VGPR j holds K = 2*(L>>4)+j  -> float2 load
// B-frag (4x16, KxN): lane L: N=L&15; VGPR j holds K = 2*(L>>4)+j
// C/D   (16x16):      lane L: N=L&15; VGPR v holds M = v + 8*(L>>4)
// ---------------------------------------------------------------------------
__global__ void gemm_xw_wmma(const float* __restrict__ X,
                             const float* __restrict__ W,
                             float* __restrict__ H,
                             int nTiles) {
    const int wave = threadIdx.x >> 5;
    const int lane = threadIdx.x & 31;
    const int tile = blockIdx.x * WAVES_PER_BLOCK + wave;
    if (tile >= nTiles) return;                 // uniform per wave: EXEC all-1s inside

    const int m   = lane & 15;                  // row within tile / col within tile
    const int kh  = lane >> 4;                  // K half-select (0 or 1)
    const float* xrow = X + (size_t)(tile * 16 + m) * DIM;

    v8f acc0 = {}, acc1 = {}, acc2 = {}, acc3 = {};

#pragma unroll
    for (int k = 0; k < DIM; k += 4) {
        // A fragment: contiguous pair K = k + 2*kh, k + 2*kh + 1
        v2f a = *(const v2f*)(xrow + k + 2 * kh);

        // B fragments for the four 16-col tiles of W
        const int kb = k + 2 * kh;
#pragma unroll
        for (int nt = 0; nt < 4; ++nt) {
            v2f b;
            b.x = W[(size_t)kb * DIM + nt * 16 + m];
            b.y = W[(size_t)(kb + 1) * DIM + nt * 16 + m];
            v8f* accp = (nt == 0) ? &acc0 : (nt == 1) ? &acc1 : (nt == 2) ? &acc2 : &acc3;
            *accp = __builtin_amdgcn_wmma_f32_16x16x4_f32(
                false, a, false, b, (short)0, *accp, false, false);
        }
    }

    // Store D: VGPR v, lane L -> row tile*16 + v + 8*kh, col nt*16 + (L&15)
#pragma unroll
    for (int nt = 0; nt < 4; ++nt) {
        const v8f acc = (nt == 0) ? acc0 : (nt == 1) ? acc1 : (nt == 2) ? acc2 : acc3;
#pragma unroll
        for (int v = 0; v < 8; ++v) {
            H[(size_t)(tile * 16 + v + 8 * kh) * DIM + nt * 16 + m] = acc[v];
        }
    }
}

// Scalar tail for rows not covered by full 16-row tiles (unused when N%16==0)
__global__ void gemm_xw_tail(const float* __restrict__ X,
                             const float* __restrict__ W,
                             float* __restrict__ H,
                             int rowStart, int nRows) {
    int idx = blockIdx.x * blockDim.x + threadIdx.x;
    if (idx >= nRows * DIM) return;
    int r = rowStart + idx / DIM;
    int c = idx % DIM;
    float s = 0.f;
#pragma unroll
    for (int k = 0; k < DIM; ++k) s += X[(size_t)r * DIM + k] * W[(size_t)k * DIM + c];
    H[(size_t)r * DIM + c] = s;
}

// ---------------------------------------------------------------------------
// Kernel 3: out[row[e]] += val[e] * H[col[e]]
// 16 threads per edge, float4 (128-bit) gather from L2-resident H,
// 4 non-returning global_atomic_add_f32 per thread.
// ---------------------------------------------------------------------------
__global__ void scatter_edges(const float* __restrict__ H,
                              const int* __restrict__ erow,
                              const int* __restrict__ ecol,
                              const float* __restrict__ eval,
                              float* __restrict__ out,
                              int E) {
    int gid = blockIdx.x * blockDim.x + threadIdx.x;
    int e = gid >> 4;
    if (e >= E) return;
    int f = (gid & 15) << 2;

    int r = erow[e];
    int c = ecol[e];
    float v = eval[e];

    float4 h = *(const float4*)(H + (size_t)c * DIM + f);
    float* op = out + (size_t)r * DIM + f;
    atomicAdd(op + 0, h.x * v);
    atomicAdd(op + 1, h.y * v);
    atomicAdd(op + 2, h.z * v);
    atomicAdd(op + 3, h.w * v);
}

// ---------------------------------------------------------------------------
extern "C" void kernel_launch(void* const* d_in, const int* in_sizes, int n_in,
                              void* d_out, int out_size, void* d_ws, size_t ws_size,
                              hipStream_t stream) {
    const float* x    = (const float*)d_in[0];   // [N, 64]
    const float* wgt  = (const float*)d_in[1];   // [64, 64]
    const float* ev   = (const float*)d_in[2];   // [E]
    const int*   erow = (const int*)d_in[3];     // [E]
    const int*   ecol = (const int*)d_in[4];     // [E]
    float*       out  = (float*)d_out;           // [N, 64]
    float*       H    = (float*)d_ws;            // [N, 64] scratch: X @ W

    const int N = in_sizes[0] / DIM;
    const int E = in_sizes[2];

    // 1) out = 0
    {
        int n4 = out_size / 4;
        int blk = (n4 + 255) / 256;
        zero_f4<<<blk, 256, 0, stream>>>((float4*)out, n4);
    }

    // 2) H = X @ W  (WMMA f32)
    {
        int nTiles = N / 16;
        if (nTiles > 0) {
            int blk = (nTiles + WAVES_PER_BLOCK - 1) / WAVES_PER_BLOCK;
            gemm_xw_wmma<<<blk, WAVES_PER_BLOCK * 32, 0, stream>>>(x, wgt, H, nTiles);
        }
        int rem = N - nTiles * 16;
        if (rem > 0) {
            int work = rem * DIM;
            gemm_xw_tail<<<(work + 255) / 256, 256, 0, stream>>>(x, wgt, H, nTiles * 16, rem);
        }
    }

    // 3) out[row] += val * H[col]
    {
        long long threads = (long long)E * 16;
        int blk = (int)((threads + 255) / 256);
        scatter_edges<<<blk, 256, 0, stream>>>(H, erow, ecol, ev, out, E);
    }
}